// IntegratedLoss_60962765799808
// MI455X (gfx1250) — compile-verified
//
#include <hip/hip_runtime.h>
#include <hip/hip_bf16.h>
#include <stdint.h>

#define DEG 0.017453292519943295f
#define BLK 128
#define GTF 20   // floats per precomputed GT record
#define NG  24   // gts per image
#define NB  2    // batch
#define NC  15   // classes
#define BETA (1.0f/9.0f)

typedef float v2f __attribute__((ext_vector_type(2)));
typedef float v8f __attribute__((ext_vector_type(8)));

// ---------- helpers ----------
__device__ __forceinline__ unsigned fkey(float f) {
  unsigned u = __float_as_uint(f);
  return (u & 0x80000000u) ? ~u : (u | 0x80000000u);
}

__device__ __forceinline__ unsigned long long shflxor64(unsigned long long v, int m) {
  unsigned lo = (unsigned)v, hi = (unsigned)(v >> 32);
  lo = __shfl_xor(lo, m, 32);
  hi = __shfl_xor(hi, m, 32);
  return ((unsigned long long)hi << 32) | (unsigned long long)lo;
}

// Full-wave (32 lane) f32 sum on the matrix unit:
// A (16x4): lane L -> VGPR0=val, VGPR1=0 ; B (4x16) = ones.
// D[m][n] = val_m + val_{m+16}; lane<16 holds D[0..7][n], lane>=16 holds D[8..15][n].
// sum of own 8 D regs + xor-16 partner = total.
__device__ __forceinline__ float wave_sum32(float v) {
  v2f a; a[0] = v;    a[1] = 0.0f;
  v2f b; b[0] = 1.0f; b[1] = 1.0f;
  v8f c = {0.f,0.f,0.f,0.f,0.f,0.f,0.f,0.f};
  v8f d = __builtin_amdgcn_wmma_f32_16x16x4_f32(false, a, false, b, (short)0, c, false, false);
  float s = ((d[0]+d[1])+(d[2]+d[3])) + ((d[4]+d[5])+(d[6]+d[7]));
  s += __shfl_xor(s, 16, 32);
  return s;
}

// per-thread polygon clip buffers live in LDS: [buf(2)][xy(2)][vert(8)][BLK]
__device__ __forceinline__ float& PB(float* sp, int buf, int xy, int v, int tid) {
  return sp[(((buf << 1) + xy) * 8 + v) * BLK + tid];
}

// Sutherland-Hodgman: clip anchor rect (CCW corners) by gt rect half-planes.
__device__ float clip_area(float* sp, int tid,
                           const float* acx, const float* acy,
                           const float* gcx, const float* gcy) {
  int cur = 0, n = 4;
  #pragma unroll
  for (int i = 0; i < 4; ++i) { PB(sp,0,0,i,tid) = acx[i]; PB(sp,0,1,i,tid) = acy[i]; }
  for (int e = 0; e < 4 && n > 0; ++e) {
    float ex0 = gcx[e],        ey0 = gcy[e];
    float ex1 = gcx[(e+1)&3],  ey1 = gcy[(e+1)&3];
    float dx = ex1 - ex0, dy = ey1 - ey0;
    int nxt = cur ^ 1, m = 0;
    float px = PB(sp,cur,0,n-1,tid), py = PB(sp,cur,1,n-1,tid);
    float dp = dx*(py - ey0) - dy*(px - ex0);
    for (int i = 0; i < n; ++i) {
      float qx = PB(sp,cur,0,i,tid), qy = PB(sp,cur,1,i,tid);
      float dq = dx*(qy - ey0) - dy*(qx - ex0);
      bool pin = dp >= 0.f, qin = dq >= 0.f;
      if (pin) { PB(sp,nxt,0,m,tid) = px; PB(sp,nxt,1,m,tid) = py; ++m; }
      if (pin != qin) {
        float den = dp - dq;
        float tt = dp / (den != 0.f ? den : 1e-30f);
        PB(sp,nxt,0,m,tid) = px + tt*(qx - px);
        PB(sp,nxt,1,m,tid) = py + tt*(qy - py); ++m;
      }
      px = qx; py = qy; dp = dq;
    }
    cur = nxt; n = m;
  }
  if (n < 3) return 0.f;
  float area = 0.f;
  float px = PB(sp,cur,0,n-1,tid), py = PB(sp,cur,1,n-1,tid);
  for (int i = 0; i < n; ++i) {
    float qx = PB(sp,cur,0,i,tid), qy = PB(sp,cur,1,i,tid);
    area += px*qy - py*qx;
    px = qx; py = qy;
  }
  return 0.5f * fabsf(area);
}

// ---------- K0: init workspace + precompute GT records ----------
__global__ void k_init(const float* __restrict__ ann, float* __restrict__ gtpre,
                       unsigned long long* __restrict__ keys, int* __restrict__ forced,
                       float* __restrict__ accum, int A) {
  int t = blockIdx.x * blockDim.x + threadIdx.x;
  if (t < NB * A) forced[t] = 0;
  if (t < 8) accum[t] = 0.0f;
  if (t < NB * NG) {
    keys[t] = 0ull;
    const float* r = ann + (size_t)t * 6;
    float cx = r[0], cy = r[1], w = r[2], h = r[3], th = r[4], lb = r[5];
    float* o = gtpre + (size_t)t * GTF;
    float ca, sa; __sincosf(th * DEG, &sa, &ca);
    o[0] = cx; o[1] = cy; o[2] = w; o[3] = h;
    o[4] = tanf(th * DEG);
    o[5] = (lb >= 0.0f) ? 1.0f : 0.0f;
    o[6] = w * h;
    o[7] = lb;
    float s = 0.5f * fmaxf(w, h);
    o[8] = cx - s; o[9] = cy - s; o[10] = cx + s; o[11] = cy + s;
    const float lx[4] = {-0.5f, 0.5f, 0.5f, -0.5f};
    const float ly[4] = {-0.5f, -0.5f, 0.5f, 0.5f};
    #pragma unroll
    for (int i = 0; i < 4; ++i) {
      o[12 + i] = cx + lx[i]*w*ca - ly[i]*h*sa;
      o[16 + i] = cy + lx[i]*w*sa + ly[i]*h*ca;
    }
  }
}

// ---------- K1: rotated overlaps, per-anchor max/argmax, per-gt packed argmax ----------
__global__ void k_overlap(const float* __restrict__ anchors, const float* __restrict__ gtpre,
                          float* __restrict__ ioumax, int* __restrict__ iouarg,
                          unsigned long long* __restrict__ keys, int A) {
  __shared__ float s_gt[NG * GTF];                 // 1.92 KB
  __shared__ float s_poly[2 * 2 * 8 * BLK];       // 16 KB clip buffers
  const int b = blockIdx.y;
  const int t = threadIdx.x;
  const int a = blockIdx.x * BLK + t;
  const bool act = (a < A);
  const int ai = act ? a : (A - 1);

  // Stage this batch's GT table into LDS via CDNA5 async-to-LDS DMA path.
  {
    unsigned long long gsrc = (unsigned long long)(uintptr_t)(gtpre + (size_t)b * NG * GTF);
    uint32_t ldsbase = (uint32_t)(uintptr_t)(void*)s_gt;
    for (int i = t; i < NG * GTF; i += BLK) {
      uint32_t lds  = ldsbase + (uint32_t)(i * 4);
      uint32_t voff = (uint32_t)(i * 4);
      asm volatile("global_load_async_to_lds_b32 %0, %1, %2"
                   :: "v"(lds), "v"(voff), "s"(gsrc) : "memory");
    }
    asm volatile("s_wait_asynccnt 0" ::: "memory");
  }
  __syncthreads();

  const float* ap = anchors + ((size_t)b * A + ai) * 5;
  float ax = ap[0], ay = ap[1], aw = ap[2], ah = ap[3], ath = ap[4];
  float ca, sa; __sincosf(ath * DEG, &sa, &ca);
  const float lx[4] = {-0.5f, 0.5f, 0.5f, -0.5f};
  const float ly[4] = {-0.5f, -0.5f, 0.5f, 0.5f};
  float acx[4], acy[4];
  #pragma unroll
  for (int i = 0; i < 4; ++i) {
    acx[i] = ax + lx[i]*aw*ca - ly[i]*ah*sa;
    acy[i] = ay + lx[i]*aw*sa + ly[i]*ah*ca;
  }
  float sq = 0.5f * fmaxf(aw, ah);
  float asx0 = ax - sq, asy0 = ay - sq, asx1 = ax + sq, asy1 = ay + sq;
  float asq_area = (2.f*sq) * (2.f*sq);
  float aarea = aw * ah;

  float best = -2.0f; int barg = 0;

  for (int g = 0; g < NG; ++g) {
    const float* G = s_gt + g * GTF;
    float valid = G[5];
    // axis-aligned square-IoU indicator gate
    float ltx = fmaxf(asx0, G[8]),  lty = fmaxf(asy0, G[9]);
    float rbx = fminf(asx1, G[10]), rby = fminf(asy1, G[11]);
    float iw = fmaxf(rbx - ltx, 0.f), ih = fmaxf(rby - lty, 0.f);
    float isq = iw * ih;
    float gsq_area = (G[10] - G[8]) * (G[11] - G[9]);
    float ind = isq / (asq_area + gsq_area - isq + 1e-9f);

    float iou;
    if (valid < 0.5f) {
      iou = -1.0f;
    } else if (ind >= 0.1f) {
      float gx[4] = {G[12], G[13], G[14], G[15]};
      float gy[4] = {G[16], G[17], G[18], G[19]};
      float inter = clip_area(s_poly, t, acx, acy, gx, gy);
      iou = inter / (aarea + G[6] - inter + 1e-9f);
    } else {
      iou = 0.0f;
    }
    if (act && iou > best) { best = iou; barg = g; }

    // per-gt argmax over anchors: monotone float key, tie -> smallest anchor idx
    unsigned long long key =
        act ? (((unsigned long long)fkey(iou) << 32) |
               (unsigned long long)(0x7FFFFFFFu - (unsigned)a))
            : 0ull;
    #pragma unroll
    for (int o = 16; o >= 1; o >>= 1) {
      unsigned long long other = shflxor64(key, o);
      key = (other > key) ? other : key;
    }
    if ((t & 31) == 0) atomicMax(&keys[b * NG + g], key);
  }

  if (act) {
    ioumax[(size_t)b * A + a] = best;
    iouarg[(size_t)b * A + a] = barg;
  }
}

// ---------- K2: force-assign gts whose best anchor IoU < 0.5 ----------
__global__ void k_force(const unsigned long long* __restrict__ keys,
                        const float* __restrict__ gtpre, int* __restrict__ forced, int A) {
  int t = threadIdx.x;
  if (t < NB * NG) {
    unsigned long long key = keys[t];
    float valid = gtpre[(size_t)t * GTF + 5];
    unsigned long long thr = ((unsigned long long)fkey(0.5f)) << 32;
    if (valid > 0.5f && key < thr) {
      unsigned a = 0x7FFFFFFFu - (unsigned)(key & 0xFFFFFFFFull);
      if (a < (unsigned)A) forced[(t / NG) * A + (int)a] = 1;
    }
  }
}

// ---------- K3: focal cls loss + smooth-L1 reg loss, WMMA wave reductions ----------
__global__ void k_loss(const float* __restrict__ cls, const float* __restrict__ reg,
                       const float* __restrict__ anchors, const float* __restrict__ gtpre,
                       const float* __restrict__ ioumax, const int* __restrict__ iouarg,
                       const int* __restrict__ forced, float* __restrict__ accum, int A) {
  const int b = blockIdx.y;
  const int t = threadIdx.x;
  const int a = blockIdx.x * BLK + t;
  const bool act = (a < A);
  const int ai = act ? a : 0;

  float im = ioumax[(size_t)b * A + ai];
  int   ag = iouarg[(size_t)b * A + ai];
  int   fo = forced[(size_t)b * A + ai];
  bool pos = act && ((im >= 0.5f) || (fo != 0));

  const float* G = gtpre + ((size_t)b * NG + ag) * GTF;
  float clsl = 0.f, regl = 0.f, np = pos ? 1.f : 0.f;

  if (act) {
    int lab = (int)G[7];
    const float* cp = cls + ((size_t)b * A + ai) * NC;
    float tneg = (im < 0.4f) ? 0.f : -1.f;  // IOU_THRES - 0.1
    #pragma unroll
    for (int c = 0; c < NC; ++c) {
      float tg = pos ? ((c == lab) ? 1.f : 0.f) : tneg;
      if (tg >= 0.f) {
        float p = fminf(fmaxf(cp[c], 1e-4f), 1.f - 1e-4f);
        float alpha = (tg == 1.f) ? 0.25f : 0.75f;
        float w = (tg == 1.f) ? (1.f - p) : p;
        float fw = alpha * w * w;
        float bce = -(tg * logf(p + 1e-6f) + (1.f - tg) * logf(1.f - p + 1e-6f));
        clsl += fw * bce;
      }
    }
  }

  if (pos) {
    const float* ap = anchors + ((size_t)b * A + ai) * 5;
    float ax = ap[0], ay = ap[1], aw = ap[2], ah = ap[3], ath = ap[4];
    float ew = fmaxf(aw, 1.f), eh = fmaxf(ah, 1.f);
    float gw = fmaxf(G[2], 1.f), gh = fmaxf(G[3], 1.f);
    float tgt[5];
    tgt[0] = 10.f * (G[0] - ax) / ew;
    tgt[1] = 10.f * (G[1] - ay) / eh;
    tgt[2] = 5.f * logf(gw / ew);
    tgt[3] = 5.f * logf(gh / eh);
    tgt[4] = 15.f * (G[4] - tanf(ath * DEG));
    const float* rp = reg + ((size_t)b * A + ai) * 5;
    #pragma unroll
    for (int j = 0; j < 5; ++j) {
      float d = fabsf(rp[j] - tgt[j]);
      regl += (d < BETA) ? (0.5f * d * d / BETA) : (d - 0.5f * BETA);
    }
  }

  // matrix-unit wave reductions (EXEC all ones here: no divergence at this point)
  float s_cls = wave_sum32(clsl);
  float s_reg = wave_sum32(regl);
  float s_np  = wave_sum32(np);
  if ((t & 31) == 0) {
    atomicAdd(&accum[b],     s_cls);
    atomicAdd(&accum[2 + b], s_reg);
    atomicAdd(&accum[4 + b], s_np);
  }
}

// ---------- K4: finalize means ----------
__global__ void k_final(const float* __restrict__ accum, float* __restrict__ out) {
  if (threadIdx.x == 0 && blockIdx.x == 0) {
    float cl = 0.f, rl = 0.f;
    #pragma unroll
    for (int b = 0; b < NB; ++b) {
      float npv = accum[4 + b];
      cl += accum[b] / fmaxf(npv, 1.f);
      float r = accum[2 + b] / fmaxf(npv * 5.f, 1.f);
      rl += (npv > 0.f) ? r : 0.f;
    }
    out[0] = cl * (1.0f / NB);
    out[1] = rl * (1.0f / NB);
  }
}

extern "C" void kernel_launch(void* const* d_in, const int* in_sizes, int n_in,
                              void* d_out, int out_size, void* d_ws, size_t ws_size,
                              hipStream_t stream) {
  const float* cls     = (const float*)d_in[0];
  const float* reg     = (const float*)d_in[1];
  const float* anchors = (const float*)d_in[2];
  const float* ann     = (const float*)d_in[3];
  const int A = in_sizes[2] / (NB * 5);   // anchors flat = B*A*5

  char* ws = (char*)d_ws;
  float*              gtpre  = (float*)ws;                                   // 2*24*20 f
  unsigned long long* keys   = (unsigned long long*)(ws + 4096);             // 48 u64
  int*                forced = (int*)(ws + 8192);                            // 2*A i32
  float*              ioumax = (float*)(ws + 8192 + (size_t)2 * A * 4);      // 2*A f32
  int*                iouarg = (int*)(ws + 8192 + (size_t)4 * A * 4);        // 2*A i32
  float*              accum  = (float*)(ws + 8192 + (size_t)6 * A * 4);      // 8 f32

  k_init<<<(NB * A + 255) / 256, 256, 0, stream>>>(ann, gtpre, keys, forced, accum, A);

  dim3 g1((A + BLK - 1) / BLK, NB);
  k_overlap<<<g1, BLK, 0, stream>>>(anchors, gtpre, ioumax, iouarg, keys, A);
  k_force<<<1, 64, 0, stream>>>(keys, gtpre, forced, A);
  k_loss<<<g1, BLK, 0, stream>>>(cls, reg, anchors, gtpre, ioumax, iouarg, forced, accum, A);
  k_final<<<1, 1, 0, stream>>>(accum, (float*)d_out);
}